// HistogramLoss_62938450756088
// MI455X (gfx1250) — compile-verified
//
#include <hip/hip_runtime.h>
#include <hip/hip_bf16.h>

#define NUM_BINS   32
#define BSAMPLES   64
#define THREADS    256
#define NWAVES     (THREADS / 32)
#define LANE_STRIDE 67   // odd stride: bank = (3*lane + bin) mod 64 -> lanes hit distinct banks
#define PRED_OFF   33

typedef __attribute__((ext_vector_type(2))) float v2f;
typedef __attribute__((ext_vector_type(8))) float v8f;

// ---------------------------------------------------------------------------
// Kernel 1: masked dual histogram (obs + pred share one mask read).
// Lane-private LDS counters -> conflict-free ds_add_u32, then 64 global
// atomics per block into per-sample count buffers in the workspace.
// ---------------------------------------------------------------------------
__global__ __launch_bounds__(THREADS) void hist_kernel(
    const float* __restrict__ obs,
    const float* __restrict__ pred,
    const unsigned char* __restrict__ mask,
    const float* __restrict__ edges,
    unsigned* __restrict__ cnt_obs,
    unsigned* __restrict__ cnt_pred,
    unsigned* __restrict__ nmask,
    int n_per_sample, int blocks_per_sample, int chunk)
{
    __shared__ unsigned hist[NWAVES * 32 * LANE_STRIDE];
    __shared__ unsigned redbuf[THREADS];

    const int tid  = threadIdx.x;
    const int wave = tid >> 5;
    const int lane = tid & 31;
    unsigned* myh = &hist[(wave * 32 + lane) * LANE_STRIDE];

    for (int i = tid; i < NWAVES * 32 * LANE_STRIDE; i += THREADS) hist[i] = 0u;
    __syncthreads();

    const int s     = blockIdx.x / blocks_per_sample;
    const int c     = blockIdx.x % blocks_per_sample;
    const int start = c * chunk;
    const int end   = min(start + chunk, n_per_sample);
    if (start >= end) return;

    const float e0  = edges[0];
    const float eN  = edges[NUM_BINS];
    const float inv = (float)NUM_BINS / (eN - e0);

    const long base = (long)s * (long)n_per_sample + (long)start;
    const float4*   o4 = (const float4*)(obs  + base);
    const float4*   p4 = (const float4*)(pred + base);
    const unsigned* m4 = (const unsigned*)(mask + base);

    const int count = end - start;
    const int nvec  = count >> 2;

    unsigned nm = 0;
    for (int i = tid; i < nvec; i += THREADS) {
        float4   xo = o4[i];
        float4   xp = p4[i];
        unsigned mm = m4[i];
        nm += __popc(mm & 0x01010101u);

        float ovals[4] = {xo.x, xo.y, xo.z, xo.w};
        float pvals[4] = {xp.x, xp.y, xp.z, xp.w};
#pragma unroll
        for (int j = 0; j < 4; ++j) {
            const unsigned mb = (mm >> (8 * j)) & 0xFFu;
            if (mb) {
                float x = ovals[j];
                if (x >= e0 && x <= eN) {
                    int b = min((int)((x - e0) * inv), NUM_BINS - 1);
                    atomicAdd(&myh[b], 1u);                    // ds_add_u32, lane-private
                }
                float y = pvals[j];
                if (y >= e0 && y <= eN) {
                    int b = min((int)((y - e0) * inv), NUM_BINS - 1);
                    atomicAdd(&myh[PRED_OFF + b], 1u);
                }
            }
        }
    }
    // ragged tail (not hit for 512*512 / 32 blocks, kept for generality)
    for (int i = (nvec << 2) + tid; i < count; i += THREADS) {
        unsigned mb = mask[base + i];
        if (mb) {
            nm++;
            float x = obs[base + i];
            if (x >= e0 && x <= eN)
                atomicAdd(&myh[min((int)((x - e0) * inv), NUM_BINS - 1)], 1u);
            float y = pred[base + i];
            if (y >= e0 && y <= eN)
                atomicAdd(&myh[PRED_OFF + min((int)((y - e0) * inv), NUM_BINS - 1)], 1u);
        }
    }
    __syncthreads();

    // reduce the 2*32 bins across NWAVES*32 lane-private copies
    for (int o = tid; o < 2 * NUM_BINS; o += THREADS) {
        const int which = o >> 5;
        const int b     = o & 31;
        unsigned sum = 0;
        for (int l = 0; l < NWAVES * 32; ++l)
            sum += hist[l * LANE_STRIDE + which * PRED_OFF + b];
        unsigned* dst = which ? cnt_pred : cnt_obs;
        atomicAdd(&dst[s * NUM_BINS + b], sum);
    }

    redbuf[tid] = nm;
    __syncthreads();
    for (int st = THREADS / 2; st > 0; st >>= 1) {
        if (tid < st) redbuf[tid] += redbuf[tid + st];
        __syncthreads();
    }
    if (tid == 0) atomicAdd(&nmask[s], redbuf[0]);
}

// ---------------------------------------------------------------------------
// Kernel 2: single-block finalize. The mean over samples (ones(1x64)@P(64x32))
// is done with V_WMMA_F32_16X16X4_F32 on wave 0 (A = ones, 16 K-steps x 2
// column slabs accumulate the 64-row column sums into C row 0).
// ---------------------------------------------------------------------------
__global__ __launch_bounds__(256) void finalize_kernel(
    const unsigned* __restrict__ cnt_obs,
    const unsigned* __restrict__ cnt_pred,
    const unsigned* __restrict__ nmask,
    const float* __restrict__ edges,
    float* __restrict__ out)
{
    __shared__ float sPo[BSAMPLES * NUM_BINS];
    __shared__ float sPp[BSAMPLES * NUM_BINS];
    __shared__ float sAvg[NUM_BINS];
    __shared__ float sW[NUM_BINS];
    __shared__ float sBW[NUM_BINS];
    __shared__ float sCE[BSAMPLES];
    __shared__ float sW2[BSAMPLES];
    __shared__ float sNorm;

    const int tid = threadIdx.x;

    // stage 1: per-sample proportions (+ zero-mask "all ones" rule, smoothing)
    if (tid < BSAMPLES) {
        const int r = tid;
        const bool zero = (nmask[r] == 0u);
        float tot = 0.f;
        for (int b = 0; b < NUM_BINS; ++b)
            tot += zero ? 1.f : (float)cnt_obs[r * NUM_BINS + b];
        tot = fmaxf(tot, 1.f);
        for (int b = 0; b < NUM_BINS; ++b) {
            float cv = zero ? 1.f : (float)cnt_obs[r * NUM_BINS + b];
            sPo[r * NUM_BINS + b] = cv / tot;
        }
        tot = 0.f;
        for (int b = 0; b < NUM_BINS; ++b)
            tot += zero ? 1.f : (float)cnt_pred[r * NUM_BINS + b];
        tot = fmaxf(tot, 1.f);
        for (int b = 0; b < NUM_BINS; ++b) {
            float cv = zero ? 1.f : (float)cnt_pred[r * NUM_BINS + b];
            sPp[r * NUM_BINS + b] = 0.95f * (cv / tot) + 0.05f / (float)NUM_BINS;
        }
    }
    __syncthreads();

    // stage 2: avg_j = mean over 64 samples of p_obs[:, j]
    if (tid < 32) {
        const int lane  = tid;
        const int n     = lane & 15;   // column within slab
        const int khalf = lane >> 4;   // which K-pair this lane half supplies
#if __has_builtin(__builtin_amdgcn_wmma_f32_16x16x4_f32)
        v2f a; a.x = 1.0f; a.y = 1.0f;                 // A = ones(16x4)
        for (int slab = 0; slab < 2; ++slab) {
            v8f cacc = {};
            for (int kb = 0; kb < 16; ++kb) {          // 16 x (K=4) covers 64 rows
                const int row0 = kb * 4 + khalf * 2;
                v2f bf;
                bf.x = sPo[(row0 + 0) * NUM_BINS + slab * 16 + n];
                bf.y = sPo[(row0 + 1) * NUM_BINS + slab * 16 + n];
                cacc = __builtin_amdgcn_wmma_f32_16x16x4_f32(
                    false, a, false, bf, (short)0, cacc, false, false);
            }
            if (lane < 16)                             // C row 0, N = lane
                sAvg[slab * 16 + lane] = cacc[0] * (1.0f / (float)BSAMPLES);
        }
#else
        float sum = 0.f;
        for (int r = 0; r < BSAMPLES; ++r) sum += sPo[r * NUM_BINS + lane];
        sAvg[lane] = sum * (1.0f / (float)BSAMPLES);
#endif
    }
    __syncthreads();

    // stage 3: class weights + midpoint bin widths
    if (tid < NUM_BINS) {
        const int j = (tid < NUM_BINS - 1) ? tid : (NUM_BINS - 2);
        const float m0 = 0.5f * (edges[j] + edges[j + 1]);
        const float m1 = 0.5f * (edges[j + 1] + edges[j + 2]);
        sBW[tid] = m1 - m0;
        sW[tid]  = 1.0f / (sAvg[tid] + 1e-3f);
    }
    __syncthreads();
    if (tid == 0) {
        float s = 0.f;
        for (int i = 0; i < NUM_BINS; ++i) s += sW[i];
        sNorm = (float)NUM_BINS / s;
    }
    __syncthreads();
    if (tid < NUM_BINS) sW[tid] *= sNorm;
    __syncthreads();

    // stage 4: per-sample CE and W2-on-CDF
    if (tid < BSAMPLES) {
        const int r = tid;
        float ce = 0.f, w2 = 0.f, co = 0.f, cp = 0.f;
        for (int b = 0; b < NUM_BINS; ++b) {
            const float po = sPo[r * NUM_BINS + b];
            const float pp = sPp[r * NUM_BINS + b];
            ce += -po * __logf(pp + 1e-8f) * sW[b];
            co += po; cp += pp;
            const float d = co - cp;
            w2 += d * d * sBW[b];
        }
        sCE[r] = ce; sW2[r] = w2;
    }
    __syncthreads();

    if (tid == 0) {
        float ce = 0.f, w2 = 0.f;
        for (int r = 0; r < BSAMPLES; ++r) { ce += sCE[r]; w2 += sW2[r]; }
        ce *= (1.0f / (float)BSAMPLES);
        w2 *= (1.0f / (float)BSAMPLES);
        out[0] = (ce + 0.1f * w2) / (float)NUM_BINS;
        out[1] = ce;
        out[2] = w2;
    }
    for (int i = tid; i < BSAMPLES * NUM_BINS; i += 256) {
        out[3 + i] = sPo[i];
        out[3 + BSAMPLES * NUM_BINS + i] = sPp[i];
    }
}

extern "C" void kernel_launch(void* const* d_in, const int* in_sizes, int n_in,
                              void* d_out, int out_size, void* d_ws, size_t ws_size,
                              hipStream_t stream) {
    const float*         obs   = (const float*)d_in[0];
    const float*         pred  = (const float*)d_in[1];
    const unsigned char* mask  = (const unsigned char*)d_in[2];   // jnp bool_ = 1 byte
    const float*         edges = (const float*)d_in[3];
    float*               out   = (float*)d_out;

    const int total        = in_sizes[0];
    const int n_per_sample = total / BSAMPLES;

    unsigned* cnt_obs  = (unsigned*)d_ws;
    unsigned* cnt_pred = cnt_obs + BSAMPLES * NUM_BINS;
    unsigned* nmask    = cnt_pred + BSAMPLES * NUM_BINS;
    const size_t zbytes = (size_t)(2 * BSAMPLES * NUM_BINS + BSAMPLES) * sizeof(unsigned);
    hipMemsetAsync(d_ws, 0, zbytes, stream);

    const int bps   = 32;                                          // blocks per sample
    int chunk = (n_per_sample + bps - 1) / bps;
    chunk = (chunk + 3) & ~3;                                      // keep float4 alignment

    hist_kernel<<<dim3(BSAMPLES * bps), THREADS, 0, stream>>>(
        obs, pred, mask, edges, cnt_obs, cnt_pred, nmask,
        n_per_sample, bps, chunk);

    finalize_kernel<<<1, 256, 0, stream>>>(cnt_obs, cnt_pred, nmask, edges, out);
}